// SelfAttention_62148176773695
// MI455X (gfx1250) — compile-verified
//
#include <hip/hip_runtime.h>
#include <hip/hip_bf16.h>

// Problem constants from the reference: B=4, C=64, N=1024, H=8, D=8.
#define BATCH 4
#define CCH   64
#define NN    1024

typedef __attribute__((ext_vector_type(2))) float v2f;
typedef __attribute__((ext_vector_type(8))) float v8f;

// ---------------------------------------------------------------------------
// Stage 1: K = Wk @ x, V = Wv @ x  (per batch), fp32 WMMA 16x16x4.
// One wave computes one 16x16 output tile via 16 chained v_wmma_f32_16x16x4_f32
// (K = 64 = 16 * 4). 2048 waves total, 4 waves per 128-thread block.
//
// VGPR layouts (CDNA5 ISA 7.12.2, wave32):
//   A (16x4, MxK):  lane = M + 16*(K>=2), vgpr = K&1
//   B (4x16, KxN):  lane = N + 16*(K>=2), vgpr = K&1   (symmetric striping)
//   C/D (16x16):    lane half selects M base (0/8), vgpr v -> M = base+v, N = lane&15
// ---------------------------------------------------------------------------
__global__ void __launch_bounds__(128)
qkv_proj_wmma(const float* __restrict__ x,
              const float* __restrict__ Wk,
              const float* __restrict__ Wv,
              float* __restrict__ Kp,
              float* __restrict__ Vp)
{
    const int lane = threadIdx.x & 31;
    const int gwave = blockIdx.x * 4 + (threadIdx.x >> 5);

    // Decode tile: [b:2][proj:1][mtile:2][ntile:6]
    const int nt = gwave & 63;
    const int mt = (gwave >> 6) & 3;
    const int pj = (gwave >> 8) & 1;
    const int b  = (gwave >> 9) & 3;

    const float* __restrict__ W   = pj ? Wv : Wk;
    float* __restrict__       Out = (pj ? Vp : Kp) + b * CCH * NN;
    const float* __restrict__ xb  = x + b * CCH * NN;

    const int m0 = mt * 16;
    const int n0 = nt * 16;
    const int lm = lane & 15;            // M for A, N for B (same lane role)
    const int kh = (lane >> 4) * 2;      // K half: 0 or 2

    v8f acc = {};                        // fp32 accumulator, starts at zero
#pragma unroll
    for (int k0 = 0; k0 < CCH; k0 += 4) {
        v2f a, bm;
        // A = Wk/Wv tile, row-major [out=64][in=64]
        a.x = W[(m0 + lm) * CCH + (k0 + kh)];
        a.y = W[(m0 + lm) * CCH + (k0 + kh + 1)];
        // B = x tile, [C][N]
        bm.x = xb[(k0 + kh) * NN + (n0 + lm)];
        bm.y = xb[(k0 + kh + 1) * NN + (n0 + lm)];
        // 8 args: (neg_a, A, neg_b, B, c_mod, C, reuse_a, reuse_b)
        acc = __builtin_amdgcn_wmma_f32_16x16x4_f32(
            false, a, false, bm, (short)0, acc, false, false);
    }

    const int ncol  = n0 + lm;
    const int mbase = m0 + (lane >> 4) * 8;
#pragma unroll
    for (int v = 0; v < 8; ++v)
        Out[(mbase + v) * NN + ncol] = acc[v];
}

// ---------------------------------------------------------------------------
// Stage 2: the softmax over keys cancels q exactly, so per channel (b,c):
//   w[m]   = exp(-k[m]*s - max_m(-k*s)) ;  out[b,c,n] = (Σ w*v) / (Σ w)  ∀n
// One 256-thread block per channel; 4 elements/thread, LDS tree reductions.
// ---------------------------------------------------------------------------
__global__ void __launch_bounds__(256)
attn_collapse(const float* __restrict__ Kp,
              const float* __restrict__ Vp,
              float* __restrict__ out)
{
    const int bc = blockIdx.x;                  // b*64 + c, 0..255
    const float* __restrict__ krow = Kp + bc * NN;
    const float* __restrict__ vrow = Vp + bc * NN;
    float* __restrict__       orow = out + bc * NN;
    const float s = 0.35355339059327373f;       // 1/sqrt(D=8)
    const int t = threadIdx.x;

    __shared__ float redA[256];
    __shared__ float redB[256];

    float kv[4];
    float lmax = -__builtin_inff();
#pragma unroll
    for (int i = 0; i < 4; ++i) {
        kv[i] = -krow[t + i * 256] * s;         // logit: -k*s (q cancels)
        lmax  = fmaxf(lmax, kv[i]);
    }

    redA[t] = lmax;
    __syncthreads();
    for (int off = 128; off > 0; off >>= 1) {
        if (t < off) redA[t] = fmaxf(redA[t], redA[t + off]);
        __syncthreads();
    }
    const float gmax = redA[0];
    __syncthreads();

    float se = 0.f, sev = 0.f;
#pragma unroll
    for (int i = 0; i < 4; ++i) {
        const float e = __expf(kv[i] - gmax);   // v_exp_f32
        se  += e;
        sev += e * vrow[t + i * 256];
    }
    redA[t] = se;
    redB[t] = sev;
    __syncthreads();
    for (int off = 128; off > 0; off >>= 1) {
        if (t < off) { redA[t] += redA[t + off]; redB[t] += redB[t + off]; }
        __syncthreads();
    }
    const float result = redB[0] / redA[0];     // constant over n

#pragma unroll
    for (int i = 0; i < 4; ++i)
        orow[t + i * 256] = result;
}

// ---------------------------------------------------------------------------
// d_in order (setup_inputs): x, Wq, Wk, Wv.  Wq is mathematically dead:
// the softmax over the key axis cancels q exactly (incl. the stabilized form).
// Workspace: Kp, Vp = 2 x [4][64][1024] fp32 = 2 MB.
// ---------------------------------------------------------------------------
extern "C" void kernel_launch(void* const* d_in, const int* in_sizes, int n_in,
                              void* d_out, int out_size, void* d_ws, size_t ws_size,
                              hipStream_t stream) {
    const float* x  = (const float*)d_in[0];
    const float* Wk = (const float*)d_in[2];
    const float* Wv = (const float*)d_in[3];
    float* out = (float*)d_out;

    float* Kp = (float*)d_ws;
    float* Vp = Kp + BATCH * CCH * NN;

    // 2048 tile-waves / 4 waves per block = 512 blocks.
    qkv_proj_wmma<<<512, 128, 0, stream>>>(x, Wk, Wv, Kp, Vp);
    // One block per (b, channel).
    attn_collapse<<<BATCH * CCH, 256, 0, stream>>>(Kp, Vp, out);
}